// TransConv_25314537243265
// MI455X (gfx1250) — compile-verified
//
#include <hip/hip_runtime.h>
#include <math.h>

// Problem constants (match reference)
#define NNODES 40000
#define NEDGES 640000
#define NLAYERS 4
#define DMODEL 128
#define NHEADS 4
#define HDIM 32
#define RSQRT_C 0.17677669529663689f   // 1/sqrt(32)

typedef __attribute__((ext_vector_type(2))) float v2f;
typedef __attribute__((ext_vector_type(8))) float v8f;

// ---- order-preserving float<->uint encoding so atomicMax(u32) gives float max ----
__device__ __forceinline__ unsigned enc_f32(float f) {
    unsigned u = __float_as_uint(f);
    return (u & 0x80000000u) ? ~u : (u | 0x80000000u);
}
__device__ __forceinline__ float dec_f32(unsigned u) {
    unsigned b = (u & 0x80000000u) ? (u ^ 0x80000000u) : ~u;
    return __uint_as_float(b);
}
#define ENC_LOWEST 0x00800000u   // enc(-FLT_MAX)

// =====================================================================
// Projection GEMM: out = h[N,128] @ W[128,128] + bias, using fp32 WMMA.
// grid.x = N/16 row tiles, grid.y = 4 selects {Wq,Wk,Wv,Ws}.
// Each block: 128 threads = 4 waves; wave w computes 16x16 output tiles
// at columns [32w, 32w+16) and [32w+16, 32w+32).
// =====================================================================
__global__ __launch_bounds__(128) void proj_gemm_kernel(
    const float* __restrict__ h,
    const float* __restrict__ Wq, const float* __restrict__ Wk,
    const float* __restrict__ Wv, const float* __restrict__ Ws,
    const float* __restrict__ bq, const float* __restrict__ bk,
    const float* __restrict__ bv, const float* __restrict__ bs,
    float* __restrict__ qo, float* __restrict__ ko,
    float* __restrict__ vo, float* __restrict__ so)
{
    __shared__ float Alds[16 * 132];   // 16 rows, stride 132 floats (bank-conflict pad)

    const int bm = blockIdx.x;
    const float* W;  const float* bias;  float* out;
    switch (blockIdx.y) {
        case 0:  W = Wq; bias = bq; out = qo; break;
        case 1:  W = Wk; bias = bk; out = ko; break;
        case 2:  W = Wv; bias = bv; out = vo; break;
        default: W = Ws; bias = bs; out = so; break;
    }

    // Stage A tile (16 x 128) into LDS, float4 coalesced.
    const float4* hrow = (const float4*)(h + (size_t)bm * 16 * DMODEL);
    for (int i = threadIdx.x; i < 16 * DMODEL / 4; i += 128) {
        int r = i >> 5;              // 32 float4 per row
        int c = (i & 31) << 2;
        float4 t = hrow[i];
        Alds[r * 132 + c + 0] = t.x;
        Alds[r * 132 + c + 1] = t.y;
        Alds[r * 132 + c + 2] = t.z;
        Alds[r * 132 + c + 3] = t.w;
    }
    __syncthreads();

    const int lane = threadIdx.x & 31;
    const int wave = threadIdx.x >> 5;
    const int m    = lane & 15;            // A row / B col within tile
    const int k0   = (lane < 16) ? 0 : 2;  // lanes 0-15: K=0,1 ; lanes 16-31: K=2,3
    const int n0   = wave * 32;            // base output column for this wave

    v8f acc0 = {};
    v8f acc1 = {};

    #pragma unroll 4
    for (int kt = 0; kt < DMODEL / 4; ++kt) {
        const int kk = kt * 4 + k0;
        v2f a;
        a.x = Alds[m * 132 + kk];
        a.y = Alds[m * 132 + kk + 1];
        v2f b0, b1;
        b0.x = W[(size_t)kk * DMODEL + n0 + m];
        b0.y = W[(size_t)(kk + 1) * DMODEL + n0 + m];
        b1.x = W[(size_t)kk * DMODEL + n0 + 16 + m];
        b1.y = W[(size_t)(kk + 1) * DMODEL + n0 + 16 + m];
        acc0 = __builtin_amdgcn_wmma_f32_16x16x4_f32(false, a, false, b0,
                                                     (short)0, acc0, false, false);
        acc1 = __builtin_amdgcn_wmma_f32_16x16x4_f32(false, a, false, b1,
                                                     (short)0, acc1, false, false);
    }

    // D layout: VGPR r holds row r (lanes 0-15) / row r+8 (lanes 16-31), col = lane%16
    const int rbase = (lane < 16) ? 0 : 8;
    const float bv0 = bias[n0 + m];
    const float bv1 = bias[n0 + 16 + m];
    float* obase = out + (size_t)bm * 16 * DMODEL;
    #pragma unroll
    for (int r = 0; r < 8; ++r) {
        const int row = rbase + r;
        obase[(size_t)row * DMODEL + n0 + m]      = acc0[r] + bv0;
        obase[(size_t)row * DMODEL + n0 + 16 + m] = acc1[r] + bv1;
    }
}

// =====================================================================
// Reset per-layer scratch: encoded max -> -FLT_MAX, denom -> 0, acc -> 0
// =====================================================================
__global__ __launch_bounds__(256) void init_kernel(
    unsigned* __restrict__ m_enc, float* __restrict__ denom,
    float* __restrict__ acc, int nNH, int nND)
{
    int gid = blockIdx.x * blockDim.x + threadIdx.x;
    if (gid < nND) acc[gid] = 0.0f;
    if (gid < nNH) { m_enc[gid] = ENC_LOWEST; denom[gid] = 0.0f; }
}

// =====================================================================
// Pass 1: alpha[e,h] = <q[dst,h,:], k[src,h,:]> / sqrt(C); atomic segment max.
// Thread per (edge, head).
// =====================================================================
__global__ __launch_bounds__(256) void edge_alpha_kernel(
    const float* __restrict__ q, const float* __restrict__ k,
    const int* __restrict__ src, const int* __restrict__ dst,
    float* __restrict__ alpha, unsigned* __restrict__ m_enc)
{
    int gid = blockIdx.x * blockDim.x + threadIdx.x;
    if (gid >= NEDGES * NHEADS) return;
    const int e  = gid >> 2;
    const int hh = gid & 3;
    const int sn = src[e];
    const int dn = dst[e];

    const float4* qr = (const float4*)(q + (size_t)dn * DMODEL + hh * HDIM);
    const float4* kr = (const float4*)(k + (size_t)sn * DMODEL + hh * HDIM);
    float sum = 0.0f;
    #pragma unroll
    for (int i = 0; i < HDIM / 4; ++i) {
        float4 a = qr[i];
        float4 b = kr[i];
        sum += a.x * b.x + a.y * b.y + a.z * b.z + a.w * b.w;
    }
    const float al = sum * RSQRT_C;
    alpha[gid] = al;
    atomicMax(&m_enc[(size_t)dn * NHEADS + hh], enc_f32(al));
}

// =====================================================================
// Pass 2: ex = exp(alpha - max); denom[dst,h] += ex; acc[dst,h,:] += ex*v[src,h,:]
// (dividing by denom is deferred to finalize -> identical math to segment softmax)
// =====================================================================
__global__ __launch_bounds__(256) void edge_accum_kernel(
    const float* __restrict__ v,
    const int* __restrict__ src, const int* __restrict__ dst,
    const float* __restrict__ alpha, const unsigned* __restrict__ m_enc,
    float* __restrict__ denom, float* __restrict__ acc)
{
    int gid = blockIdx.x * blockDim.x + threadIdx.x;
    if (gid >= NEDGES * NHEADS) return;
    const int e  = gid >> 2;
    const int hh = gid & 3;
    const int sn = src[e];
    const int dn = dst[e];

    const float mm = dec_f32(m_enc[(size_t)dn * NHEADS + hh]);
    const float ex = __expf(alpha[gid] - mm);
    unsafeAtomicAdd(&denom[(size_t)dn * NHEADS + hh], ex);

    const float4* vr = (const float4*)(v + (size_t)sn * DMODEL + hh * HDIM);
    float* orow = acc + (size_t)dn * DMODEL + hh * HDIM;
    #pragma unroll
    for (int i = 0; i < HDIM / 4; ++i) {
        float4 vv = vr[i];
        unsafeAtomicAdd(orow + i * 4 + 0, ex * vv.x);
        unsafeAtomicAdd(orow + i * 4 + 1, ex * vv.y);
        unsafeAtomicAdd(orow + i * 4 + 2, ex * vv.z);
        unsafeAtomicAdd(orow + i * 4 + 3, ex * vv.w);
    }
}

// =====================================================================
// Finalize: h_next = elu(acc/denom + skip)
// =====================================================================
__global__ __launch_bounds__(256) void finalize_kernel(
    const float* __restrict__ acc, const float* __restrict__ denom,
    const float* __restrict__ skip, float* __restrict__ hnext)
{
    int gid = blockIdx.x * blockDim.x + threadIdx.x;
    if (gid >= NNODES * DMODEL) return;
    const int n  = gid >> 7;
    const int j  = gid & 127;
    const int hh = j >> 5;
    const float d = denom[(size_t)n * NHEADS + hh];
    float agg = acc[gid];
    agg = (d > 0.0f) ? (agg / d) : 0.0f;
    const float o = agg + skip[gid];
    hnext[gid] = (o > 0.0f) ? o : expm1f(o);
}

// =====================================================================
extern "C" void kernel_launch(void* const* d_in, const int* in_sizes, int n_in,
                              void* d_out, int out_size, void* d_ws, size_t ws_size,
                              hipStream_t stream)
{
    const float* x  = (const float*)d_in[0];
    const int*   ei = (const int*)d_in[1];      // [2,E]: row 0 = src, row 1 = dst
    const float* Wq = (const float*)d_in[2];
    const float* bq = (const float*)d_in[3];
    const float* Wk = (const float*)d_in[4];
    const float* bk = (const float*)d_in[5];
    const float* Wv = (const float*)d_in[6];
    const float* bv = (const float*)d_in[7];
    const float* Ws = (const float*)d_in[8];
    const float* bs = (const float*)d_in[9];
    float* out = (float*)d_out;

    const int* src = ei;
    const int* dst = ei + NEDGES;

    const size_t ND = (size_t)NNODES * DMODEL;
    const size_t NH = (size_t)NNODES * NHEADS;
    const size_t EH = (size_t)NEDGES * NHEADS;

    float*    ws     = (float*)d_ws;
    float*    qb     = ws;
    float*    kb     = qb + ND;
    float*    vb     = kb + ND;
    float*    sb     = vb + ND;
    float*    buf0   = sb + ND;
    float*    buf1   = buf0 + ND;
    float*    accb   = buf1 + ND;
    float*    alphab = accb + ND;
    unsigned* m_enc  = (unsigned*)(alphab + EH);
    float*    denomb = (float*)(m_enc + NH);

    const int blk = 256;
    const int gEH = (int)((EH + blk - 1) / blk);
    const int gND = (int)((ND + blk - 1) / blk);

    for (int layer = 0; layer < NLAYERS; ++layer) {
        const float* hcur = (layer == 0) ? x : ((layer & 1) ? buf1 : buf0);
        float* hnext = (layer == NLAYERS - 1) ? out : ((layer & 1) ? buf0 : buf1);

        const size_t woff = (size_t)layer * DMODEL * DMODEL;
        const size_t boff = (size_t)layer * DMODEL;

        proj_gemm_kernel<<<dim3(NNODES / 16, 4), 128, 0, stream>>>(
            hcur,
            Wq + woff, Wk + woff, Wv + woff, Ws + woff,
            bq + boff, bk + boff, bv + boff, bs + boff,
            qb, kb, vb, sb);

        init_kernel<<<gND, blk, 0, stream>>>(m_enc, denomb, accb, (int)NH, (int)ND);

        edge_alpha_kernel<<<gEH, blk, 0, stream>>>(qb, kb, src, dst, alphab, m_enc);

        edge_accum_kernel<<<gEH, blk, 0, stream>>>(vb, src, dst, alphab, m_enc,
                                                   denomb, accb);

        finalize_kernel<<<gND, blk, 0, stream>>>(accb, denomb, sb, hnext);
    }
}